// SpatialDistangleLoss_9740985827724
// MI455X (gfx1250) — compile-verified
//
#include <hip/hip_runtime.h>
#include <hip/hip_bf16.h>
#include <stdint.h>

// ---------------------------------------------------------------------------
// SpatialDistangleLoss for MI455X (gfx1250).
// loss = sum_e a_e * (box5x5(a)_e - a_e) / (25*H*W*B),  a = |y|
// Memory-bound (94.4 MB -> 1 scalar). Data movement uses the CDNA5 Tensor
// Data Mover (tensor_load_to_lds + s_wait_tensorcnt); math is a separable
// 5x5 box filter entirely in LDS; one f32 atomic per block.
// ---------------------------------------------------------------------------

typedef __attribute__((ext_vector_type(4))) unsigned int u32x4;
typedef __attribute__((ext_vector_type(4))) int          i32x4;
typedef __attribute__((ext_vector_type(8))) int          i32x8;

#define HH     96
#define WW     96
#define PLANE  (HH * WW)        // 9216 floats = 36 KB
#define NT     256              // 8 wave32
#define ITERS  (PLANE / NT)     // 36 elements per thread
#define BB     8                // batch (fixed by reference)

__global__ __launch_bounds__(NT) void
sdl_kernel(const float* __restrict__ y, float* __restrict__ out)
{
    extern __shared__ float smem[];
    float* raw = smem;              // [0, PLANE)      raw plane (TDM destination)
    float* rs  = smem + PLANE;      // [PLANE, 2*PLANE) row box-sums of |raw|
    float* red = smem + 2 * PLANE;  // 8 per-wave partials

    const int tid   = threadIdx.x;
    const int plane = blockIdx.x;                       // (b,c) plane index
    const float* gsrc = y + (size_t)plane * PLANE;

    // ---------------- Phase 0: DMA plane into LDS via the TDM ----------------
#if __has_builtin(__builtin_amdgcn_tensor_load_to_lds)
    if ((tid >> 5) == 0) {                              // wave 0 issues the DMA
        const uint64_t ga = (uint64_t)(uintptr_t)gsrc;
        const uint32_t la = (uint32_t)(uintptr_t)raw;   // LDS byte address (flat low 32)

        u32x4 g0;
        g0[0] = 1u;                                         // count=1, user D#
        g0[1] = la;                                         // lds_addr
        g0[2] = (uint32_t)ga;                               // global_addr[31:0]
        g0[3] = (uint32_t)((ga >> 32) & 0x01FFFFFFu)        // global_addr[56:32]
              | (2u << 30);                                 // type = 2 ("image")

        i32x8 g1;
        g1[0] = (int)(2u << 16);                            // data_size = 4 bytes
        g1[1] = (int)(((unsigned)PLANE & 0xFFFFu) << 16);   // tensor_dim0[15:0]
        g1[2] = (int)(((unsigned)PLANE >> 16) | (1u << 16));// dim0[31:16] | tensor_dim1=1
        g1[3] = (int)((unsigned)PLANE << 16);               // tile_dim0 = 9216 (1-D tile)
        g1[4] = 1;                                          // tile_dim1 = 1, tile_dim2 = 0
        g1[5] = PLANE;                                      // tensor_dim0_stride[31:0]
        g1[6] = 0;                                          // stride hi / dim1_stride lo
        g1[7] = 0;

        i32x4 gz = (i32x4)(0);                              // groups 2/3 unused (<=2D)
#if defined(__clang_major__) && (__clang_major__ >= 23)
        i32x8 gz8 = (i32x8)(0);
        __builtin_amdgcn_tensor_load_to_lds(g0, g1, gz, gz, gz8, 0);
#else
        __builtin_amdgcn_tensor_load_to_lds(g0, g1, gz, gz, 0);
#endif
#if __has_builtin(__builtin_amdgcn_s_wait_tensorcnt)
        __builtin_amdgcn_s_wait_tensorcnt(0);
#else
        asm volatile("s_wait_tensorcnt 0x0" ::: "memory");
#endif
    }
#else
    // Fallback path if the TDM builtin is unavailable: cooperative 128-bit copy.
    {
        const float4* s4 = (const float4*)gsrc;
        float4*       d4 = (float4*)raw;
        for (int i = tid; i < PLANE / 4; i += NT) d4[i] = s4[i];
    }
#endif
    __syncthreads();

    // ------------- Phase 1: horizontal 5-tap box-sum of |raw| ---------------
    // (fabsf folds into VALU source modifiers on the adds)
    for (int it = 0; it < ITERS; ++it) {
        const int e = tid + it * NT;
        const int h = e / WW;
        const int w = e - h * WW;
        const float* row = raw + h * WW;
        float s = fabsf(row[w]);
        if (w >= 1)      s += fabsf(row[w - 1]);
        if (w >= 2)      s += fabsf(row[w - 2]);
        if (w <= WW - 2) s += fabsf(row[w + 1]);
        if (w <= WW - 3) s += fabsf(row[w + 2]);
        rs[e] = s;
    }
    __syncthreads();

    // ------ Phase 2: vertical 5-tap sum + dot product a*(box25(a)-a) --------
    float partial = 0.0f;
    for (int it = 0; it < ITERS; ++it) {
        const int e = tid + it * NT;
        const int h = e / WW;
        float S = rs[e];
        if (h >= 1)      S += rs[e - WW];
        if (h >= 2)      S += rs[e - 2 * WW];
        if (h <= HH - 2) S += rs[e + WW];
        if (h <= HH - 3) S += rs[e + 2 * WW];
        const float a = fabsf(raw[e]);
        partial = fmaf(a, S - a, partial);
    }

    // ---------------- Phase 3: hierarchical reduction -----------------------
    #pragma unroll
    for (int off = 16; off > 0; off >>= 1)
        partial += __shfl_down(partial, off, 32);           // wave32 reduction
    if ((tid & 31) == 0) red[tid >> 5] = partial;
    __syncthreads();

    if (tid == 0) {
        float s = 0.0f;
        #pragma unroll
        for (int wv = 0; wv < NT / 32; ++wv) s += red[wv];
        const float scale = 1.0f / (25.0f * (float)(HH * WW) * (float)BB);
        atomicAdd(out, s * scale);                          // global_atomic_add_f32
    }
}

// ---------------------------------------------------------------------------
extern "C" void kernel_launch(void* const* d_in, const int* in_sizes, int n_in,
                              void* d_out, int out_size, void* d_ws, size_t ws_size,
                              hipStream_t stream)
{
    (void)n_in; (void)d_ws; (void)ws_size; (void)out_size;
    const float* y   = (const float*)d_in[0];
    float*       out = (float*)d_out;

    // Zero the scalar accumulator (graph-capture safe).
    hipMemsetAsync(out, 0, sizeof(float), stream);

    const int nplanes = in_sizes[0] / PLANE;                // 8*320 = 2560
    const size_t shmem = (size_t)(2 * PLANE + 8) * sizeof(float); // 73,760 B

    sdl_kernel<<<dim3(nplanes), dim3(NT), shmem, stream>>>(y, out);
}